// ObjectEncoder_39307540693438
// MI455X (gfx1250) — compile-verified
//
#include <hip/hip_runtime.h>

// CDNA5 / gfx1250 ObjectEncoder.
//
// Algorithmic restructure: aff/mat vocab are 16 each -> the whole per-part
// MLP (relu(W_part @ concat + b)) has only 256 distinct outputs. Precompute
// a 256x256 table; max-pool becomes a table gather. task/obj/state slices of
// W_obj similarly become small additive tables. Only real GEMM left:
// pooled[B,256] @ W_obj[:,192:448]^T  -> v_wmma_f32_16x16x4_f32.
//
// Round 2: uniform (block-level) tail guard -> straight-line epilogue;
// B-matrix repacked so one global_load_b128 feeds two WMMA k-groups;
// float4 table gather in stage A.

typedef __attribute__((ext_vector_type(2))) float v2f;
typedef __attribute__((ext_vector_type(8))) float v8f;

constexpr int PART_D = 256;   // part encoder width == pooled dim
constexpr int OUT_D  = 256;   // object encoder width
constexpr int TB     = 64;    // batch rows per block

// Workspace layout (float offsets)
constexpr int WS_PT = 0;                 // part_table   [256][256]
constexpr int WS_WI = WS_PT + 256 * 256; // W_pool^T interleaved (see below)
constexpr int WS_TT = WS_WI + 256 * 256; // task contrib [16][256]  (+ b_obj)
constexpr int WS_OT = WS_TT + 16 * 256;  // obj contrib  [64][256]
constexpr int WS_ST = WS_OT + 64 * 256;  // state contrib[32][256]
// total = 159744 floats = 624 KB of d_ws
//
// WI layout: float4 element e = (g2*2 + h)*256 + n, component q, holds
// W_pool^T row k = 8*g2 + 2*q + h, column n  (h = lane-half parity).
// One b128 per lane covers the B fragments of two 16x16x4 k-groups.

// ---------------------------------------------------------------------------
// Table precompute. 624 blocks x 256 threads, thread = output column.
// ---------------------------------------------------------------------------
__global__ __launch_bounds__(256) void obj_enc_prep_kernel(
    const float* __restrict__ aff_emb,  const float* __restrict__ mat_emb,
    const float* __restrict__ task_emb, const float* __restrict__ obj_emb,
    const float* __restrict__ state_emb,
    const float* __restrict__ W_part,   const float* __restrict__ b_part,
    const float* __restrict__ W_obj,    const float* __restrict__ b_obj,
    float* __restrict__ ws)
{
    float* PT = ws + WS_PT;
    float* WI = ws + WS_WI;
    float* TT = ws + WS_TT;
    float* OT = ws + WS_OT;
    float* ST = ws + WS_ST;

    const int bid = blockIdx.x;
    const int t   = threadIdx.x;   // output column 0..255

    if (bid < 256) {
        // part_table[combo][t] = relu(W_part[t,:] . concat(aff[a], mat[m]) + b)
        const int a = bid >> 4, m = bid & 15;
        const float* w = W_part + t * 128;
        float s = b_part[t];
        #pragma unroll 4
        for (int k = 0; k < 64; ++k) s += w[k]      * aff_emb[a * 64 + k];
        #pragma unroll 4
        for (int k = 0; k < 64; ++k) s += w[64 + k] * mat_emb[m * 64 + k];
        PT[bid * PART_D + t] = fmaxf(s, 0.0f);
    } else if (bid < 512) {
        // interleaved k-major pool weights: WI[e].q = W_obj[n, 192+k]
        const int k  = bid - 256;
        const int g2 = k >> 3;
        const int h  = k & 1;
        const int q  = (k & 7) >> 1;
        WI[(((g2 * 2 + h) * 256) + t) * 4 + q] = W_obj[t * 448 + 192 + k];
    } else if (bid < 528) {
        // task contribution (+ bias folded in once here)
        const int v = bid - 512;
        float s = b_obj[t];
        #pragma unroll 4
        for (int k = 0; k < 64; ++k) s += W_obj[t * 448 + k] * task_emb[v * 64 + k];
        TT[v * OUT_D + t] = s;
    } else if (bid < 592) {
        const int v = bid - 528;
        float s = 0.0f;
        #pragma unroll 4
        for (int k = 0; k < 64; ++k) s += W_obj[t * 448 + 64 + k] * obj_emb[v * 64 + k];
        OT[v * OUT_D + t] = s;
    } else {
        const int v = bid - 592;
        float s = 0.0f;
        #pragma unroll 4
        for (int k = 0; k < 64; ++k) s += W_obj[t * 448 + 128 + k] * state_emb[v * 64 + k];
        ST[v * OUT_D + t] = s;
    }
}

// ---------------------------------------------------------------------------
// Main kernel: gather+maxpool into LDS, then WMMA fp32 GEMM + fused epilogue.
// 256 threads = 8 wave32. Wave w: M-tile = w&3 (16 rows), N-half = w>>2.
// ---------------------------------------------------------------------------
__global__ __launch_bounds__(256) void obj_enc_main_kernel(
    const int* __restrict__ aff_idx,  const int* __restrict__ mat_idx,
    const int* __restrict__ task_idx, const int* __restrict__ obj_idx,
    const int* __restrict__ state_idx,
    const float* __restrict__ ws, float* __restrict__ out, int B)
{
    const float4* PT4 = (const float4*)(ws + WS_PT);
    const float4* WI4 = (const float4*)(ws + WS_WI);
    const float4* TT4 = (const float4*)(ws + WS_TT);
    const float4* OT4 = (const float4*)(ws + WS_OT);
    const float4* ST4 = (const float4*)(ws + WS_ST);

    // stride 260 dwords: 260 % 64 == 4 -> A-fragment reads (16 lanes, 16 rows,
    // same k) hit 16 distinct LDS banks; 260*4 and 4-float columns stay 16B
    // aligned for b128 stores.
    constexpr int PSTRIDE = 260;
    __shared__ float pooled[TB][PSTRIDE];   // max-pooled part encodings
    __shared__ float basev[TB][OUT_D];      // task+obj+state contribs (+bias)
    __shared__ int   combo[TB][8];
    __shared__ int   cat3[TB][4];

    const int tid     = threadIdx.x;
    const int rowbase = blockIdx.x * TB;

    // ---- Stage A0: indices -> LDS ------------------------------------------
    for (int i = tid; i < TB * 8; i += 256) {
        const int r = i >> 3, p = i & 7;
        const int b = min(rowbase + r, B - 1);
        combo[r][p] = aff_idx[b * 8 + p] * 16 + mat_idx[b * 8 + p];
    }
    if (tid < TB) {
        const int b = min(rowbase + tid, B - 1);
        cat3[tid][0] = task_idx[b];
        cat3[tid][1] = obj_idx[b];
        cat3[tid][2] = state_idx[b];
    }
    __syncthreads();

    // ---- Stage A1: float4 table gather + max-pool --------------------------
    // 64 threads span one 256-wide row; 256 threads do 4 rows per pass.
    {
        const int r0 = tid >> 6;          // 0..3 (uniform within a wave)
        const int cq = tid & 63;          // float4 column index
        for (int r = r0; r < TB; r += 4) {
            float4 m = PT4[combo[r][0] * 64 + cq];
            #pragma unroll
            for (int p = 1; p < 8; ++p) {
                const float4 t = PT4[combo[r][p] * 64 + cq];
                m.x = fmaxf(m.x, t.x); m.y = fmaxf(m.y, t.y);
                m.z = fmaxf(m.z, t.z); m.w = fmaxf(m.w, t.w);
            }
            *(float4*)&pooled[r][cq * 4] = m;

            const float4 a = TT4[cat3[r][0] * 64 + cq];
            const float4 b = OT4[cat3[r][1] * 64 + cq];
            const float4 c = ST4[cat3[r][2] * 64 + cq];
            float4 s;
            s.x = a.x + b.x + c.x; s.y = a.y + b.y + c.y;
            s.z = a.z + b.z + c.z; s.w = a.w + b.w + c.w;
            *(float4*)&basev[r][cq * 4] = s;
        }
    }
    __syncthreads();

    // ---- Stage B: pooled[16x256] @ W_pool^T -> 16x128 per wave via WMMA ----
    const int wave  = tid >> 5;
    const int lane  = tid & 31;
    const int mtile = wave & 3;     // which 16-row tile of the 64-row block
    const int ngrp  = wave >> 2;    // which 128-col half of the 256 outputs
    const int col   = lane & 15;
    const int half  = lane >> 4;    // 0: lanes 0-15, 1: lanes 16-31

    v8f acc[8] = {};                // 8 N-tiles x 16x16 f32 accumulators

    // A fragment (16x4 f32): lanes 0-15 carry M=0..15; VGPR0/1 = K, K+1;
    // lane-half selects K base (0 or 2).  B fragment (4x16 f32): row K striped
    // across lanes 0-15 (N), lane-half selects K row parity; VGPR1 = K+2.
    const int arow = mtile * 16 + col;
    for (int g2 = 0; g2 < PART_D / 8; ++g2) {      // 8 k-values per iteration
        const int kb = g2 * 8;
        v2f a0, a1;
        a0.x = pooled[arow][kb + half * 2];
        a0.y = pooled[arow][kb + half * 2 + 1];
        a1.x = pooled[arow][kb + 4 + half * 2];
        a1.y = pooled[arow][kb + 4 + half * 2 + 1];

        float4 bq[8];
        #pragma unroll
        for (int nt = 0; nt < 8; ++nt) {
            const int n = (ngrp * 8 + nt) * 16 + col;
            bq[nt] = WI4[(g2 * 2 + half) * 256 + n];   // one b128 = 2 B frags
        }
        #pragma unroll
        for (int nt = 0; nt < 8; ++nt) {
            v2f b0; b0.x = bq[nt].x; b0.y = bq[nt].y;
            acc[nt] = __builtin_amdgcn_wmma_f32_16x16x4_f32(
                false, a0, false, b0, (short)0, acc[nt], false, false);
        }
        #pragma unroll
        for (int nt = 0; nt < 8; ++nt) {
            v2f b1; b1.x = bq[nt].z; b1.y = bq[nt].w;
            acc[nt] = __builtin_amdgcn_wmma_f32_16x16x4_f32(
                false, a1, false, b1, (short)0, acc[nt], false, false);
        }
    }

    // ---- Epilogue: + base, relu, store. Uniform tail guard -----------------
    // C/D layout: VGPR j, lanes 0-15 -> (M=j, N=lane); lanes 16-31 -> (M=j+8).
    if (rowbase + TB <= B) {
        // fast path: no per-element guards -> straight-line clauseable stores
        #pragma unroll
        for (int nt = 0; nt < 8; ++nt) {
            const int ncol = (ngrp * 8 + nt) * 16 + col;
            #pragma unroll
            for (int j = 0; j < 8; ++j) {
                const int r = mtile * 16 + j + half * 8;
                const float v = fmaxf(acc[nt][j] + basev[r][ncol], 0.0f);
                out[(rowbase + r) * OUT_D + ncol] = v;
            }
        }
    } else {
        #pragma unroll
        for (int nt = 0; nt < 8; ++nt) {
            const int ncol = (ngrp * 8 + nt) * 16 + col;
            #pragma unroll
            for (int j = 0; j < 8; ++j) {
                const int r = mtile * 16 + j + half * 8;
                const float v = fmaxf(acc[nt][j] + basev[r][ncol], 0.0f);
                if (rowbase + r < B) out[(rowbase + r) * OUT_D + ncol] = v;
            }
        }
    }
}

// ---------------------------------------------------------------------------
extern "C" void kernel_launch(void* const* d_in, const int* in_sizes, int n_in,
                              void* d_out, int out_size, void* d_ws, size_t ws_size,
                              hipStream_t stream) {
    const int*   aff_idx   = (const int*)d_in[0];
    const int*   mat_idx   = (const int*)d_in[1];
    const int*   task_idx  = (const int*)d_in[2];
    const int*   obj_idx   = (const int*)d_in[3];
    const int*   state_idx = (const int*)d_in[4];
    const float* aff_emb   = (const float*)d_in[5];
    const float* mat_emb   = (const float*)d_in[6];
    const float* task_emb  = (const float*)d_in[7];
    const float* obj_emb   = (const float*)d_in[8];
    const float* state_emb = (const float*)d_in[9];
    const float* W_part    = (const float*)d_in[10];
    const float* b_part    = (const float*)d_in[11];
    const float* W_obj     = (const float*)d_in[12];
    const float* b_obj     = (const float*)d_in[13];

    float* out = (float*)d_out;
    float* ws  = (float*)d_ws;
    const int B = in_sizes[2];   // task_idx is [B]

    obj_enc_prep_kernel<<<624, 256, 0, stream>>>(
        aff_emb, mat_emb, task_emb, obj_emb, state_emb,
        W_part, b_part, W_obj, b_obj, ws);

    const int nblocks = (B + TB - 1) / TB;
    obj_enc_main_kernel<<<nblocks, 256, 0, stream>>>(
        aff_idx, mat_idx, task_idx, obj_idx, state_idx, ws, out, B);
}